// SSDTarget_10539849744793
// MI455X (gfx1250) — compile-verified
//
#include <hip/hip_runtime.h>

// SSD anchor-target assignment for MI455X (gfx1250, wave32).
// Layout: d_out = [deltas (B,A,4) | class_ids (B,A) | anchors_tag (B,A)], all f32.
//
// Per block: 1024 anchors of one batch (256 thr x 4 anchors).
//  1) Stage gt_boxes/gt_class_ids batch slice into LDS with
//     global_load_async_to_lds_b32 (CDNA5 async path, ASYNCcnt).
//  2) Wave 0 compacts valid GT boxes (ballot_w32 prefix) preserving order.
//  3) Hot loop: division-free IoU argmax via cross-multiplied fractions.
//  4) Epilogue: 3 divides + 4 v_log_f32 per anchor, b128 delta store.

#define THREADS 256
#define APT 4            // anchors per thread
#define GMAX 64          // G in reference setup

__global__ __launch_bounds__(THREADS)
void ssd_target_gfx1250(const float* __restrict__ gt_boxes,   // (B,G,5)
                        const int*   __restrict__ gt_cls,     // (B,G,2)
                        const float* __restrict__ anchors,    // (A,4)
                        float* __restrict__ out,
                        int A, int B)
{
    __shared__ float s_raw[GMAX * 5];
    __shared__ int   s_cls[GMAX * 2];
    __shared__ __align__(16) float s_slot[GMAX * 8]; // y1,x1,y2,x2,area,cls,pad,pad
    __shared__ int   s_cnt;

    const int tid = threadIdx.x;
    const int b   = blockIdx.y;
    const long long aBase = (long long)blockIdx.x * (THREADS * APT);

    // ---- stage GT slice into LDS via async-to-LDS (gfx1250) ----
    {
        const float* gsrc = gt_boxes + (size_t)b * (GMAX * 5);
        for (int i = tid; i < GMAX * 5; i += THREADS) {
            unsigned l = (unsigned)(size_t)&s_raw[i];             // low 32b of flat = LDS offset
            unsigned long long ga = (unsigned long long)(size_t)(gsrc + i);
            asm volatile("global_load_async_to_lds_b32 %0, %1, off"
                         :: "v"(l), "v"(ga) : "memory");
        }
        const int* csrc = gt_cls + (size_t)b * (GMAX * 2);
        for (int i = tid; i < GMAX * 2; i += THREADS) {
            unsigned l = (unsigned)(size_t)&s_cls[i];
            unsigned long long ga = (unsigned long long)(size_t)(csrc + i);
            asm volatile("global_load_async_to_lds_b32 %0, %1, off"
                         :: "v"(l), "v"(ga) : "memory");
        }
        asm volatile("s_wait_asynccnt 0" ::: "memory");
    }
    __syncthreads();

    // ---- wave-0 order-preserving compaction of valid GT boxes ----
    if (tid < 32) {
        const int lane = tid;
        int cnt = 0;
#pragma unroll
        for (int r = 0; r < 2; ++r) {
            const int g = lane + r * 32;
            float y1 = s_raw[g * 5 + 0];
            float x1 = s_raw[g * 5 + 1];
            float y2 = s_raw[g * 5 + 2];
            float x2 = s_raw[g * 5 + 3];
            bool valid = (s_raw[g * 5 + 4] != 0.0f);
            unsigned m = __builtin_amdgcn_ballot_w32(valid);
            int pos = cnt + __popc(m & ((1u << lane) - 1u));
            if (valid) {
                s_slot[pos * 8 + 0] = y1;
                s_slot[pos * 8 + 1] = x1;
                s_slot[pos * 8 + 2] = y2;
                s_slot[pos * 8 + 3] = x2;
                s_slot[pos * 8 + 4] = (x2 - x1) * (y2 - y1);
                s_slot[pos * 8 + 5] = (float)s_cls[g * 2];
                s_slot[pos * 8 + 6] = 0.0f;
                s_slot[pos * 8 + 7] = 0.0f;
            }
            cnt += __popc(m);
        }
        if (lane == 0) {
            if (cnt == 0) { // all invalid -> argmax==0 semantics: use gt[0]
                float y1 = s_raw[0], x1 = s_raw[1], y2 = s_raw[2], x2 = s_raw[3];
                s_slot[0] = y1; s_slot[1] = x1; s_slot[2] = y2; s_slot[3] = x2;
                s_slot[4] = (x2 - x1) * (y2 - y1);
                s_slot[5] = (float)s_cls[0];
                s_slot[6] = 0.0f; s_slot[7] = 0.0f;
            }
            s_cnt = cnt;
        }
    }
    __syncthreads();

    // ---- load 4 anchors/thread (coalesced b128) ----
    float ay1[APT], ax1[APT], ay2[APT], ax2[APT], areaA[APT];
    bool  ok[APT];
#pragma unroll
    for (int k = 0; k < APT; ++k) {
        long long a = aBase + tid + k * THREADS;
        ok[k] = (a < A);
        float4 an = ok[k] ? ((const float4*)anchors)[a] : make_float4(0.f, 0.f, 1.f, 1.f);
        ay1[k] = an.x; ax1[k] = an.y; ay2[k] = an.z; ax2[k] = an.w;
        areaA[k] = (an.w - an.y) * (an.z - an.x);
    }

    float bn[APT], bd[APT];
    int   bs[APT];
#pragma unroll
    for (int k = 0; k < APT; ++k) { bn[k] = -1.0f; bd[k] = 1.0f; bs[k] = 0; }

    // ---- division-free IoU argmax over compacted GT ----
    const int cnt = s_cnt;
    const float4* slots = (const float4*)s_slot;
    for (int g = 0; g < cnt; ++g) {
        float4 c    = slots[g * 2];        // broadcast ds_load_b128: y1,x1,y2,x2
        float areaG = s_slot[g * 8 + 4];
#pragma unroll
        for (int k = 0; k < APT; ++k) {
            float iw = fminf(ax2[k], c.w) - fmaxf(ax1[k], c.y);
            float ih = fminf(ay2[k], c.z) - fmaxf(ay1[k], c.x);
            iw = fmaxf(iw, 0.0f);
            ih = fmaxf(ih, 0.0f);
            float inter = iw * ih;
            float den   = areaG + areaA[k] - inter;   // union > 0 for valid boxes
            bool upd = inter * bd[k] > bn[k] * den;   // inter/den > bn/bd (strict: keep first max)
            bn[k] = upd ? inter : bn[k];
            bd[k] = upd ? den   : bd[k];
            bs[k] = upd ? g     : bs[k];
        }
    }

    // ---- epilogue: tag, class, deltas ----
    float* outD = out;
    float* outC = out + (size_t)B * A * 4;
    float* outT = outC + (size_t)B * A;
    const size_t rowBase = (size_t)b * A;
    const float LOG2_5 = 0.6931471805599453f * 5.0f;  // ln(2) / BBOX_STD[2,3]
#pragma unroll
    for (int k = 0; k < APT; ++k) {
        if (!ok[k]) continue;
        long long a = aBase + tid + k * THREADS;
        float iou = bn[k] / bd[k];
        float tag = (iou >= 0.5f) ? 1.0f : ((iou < 0.4f) ? -1.0f : 0.0f);
        int s = bs[k];
        float4 m  = slots[s * 2];
        float cls = s_slot[s * 8 + 5];
        float h  = ay2[k] - ay1[k];
        float w  = ax2[k] - ax1[k];
        float cy = (ay2[k] + ay1[k]) * 0.5f;
        float cx = (ax2[k] + ax1[k]) * 0.5f;
        float gh  = m.z - m.x;
        float gw  = m.w - m.y;
        float gcy = (m.z + m.x) * 0.5f;
        float gcx = (m.w + m.y) * 0.5f;
        float dy = (gcy - cy) / h * 10.0f;            // /BBOX_STD[0]
        float dx = (gcx - cx) / w * 10.0f;            // /BBOX_STD[1]
        float dh = (__log2f(gh) - __log2f(h)) * LOG2_5;
        float dw = (__log2f(gw) - __log2f(w)) * LOG2_5;
        size_t idx = rowBase + (size_t)a;
        ((float4*)outD)[idx] = make_float4(dy, dx, dh, dw);
        outC[idx] = cls;
        outT[idx] = tag;
    }
}

extern "C" void kernel_launch(void* const* d_in, const int* in_sizes, int n_in,
                              void* d_out, int out_size, void* d_ws, size_t ws_size,
                              hipStream_t stream) {
    const float* gt_boxes = (const float*)d_in[0];   // (B,G,5) f32
    const int*   gt_cls   = (const int*)d_in[1];     // (B,G,2) i32
    const float* anchors  = (const float*)d_in[2];   // (A,4)   f32

    const int A = in_sizes[2] / 4;
    const int B = in_sizes[0] / (GMAX * 5);          // reference: B=16, G=64

    dim3 grid((A + THREADS * APT - 1) / (THREADS * APT), B);
    ssd_target_gfx1250<<<grid, THREADS, 0, stream>>>(gt_boxes, gt_cls, anchors,
                                                     (float*)d_out, A, B);
}